// ChamferLoss_11630771438180
// MI455X (gfx1250) — compile-verified
//
#include <hip/hip_runtime.h>

typedef __attribute__((ext_vector_type(2))) float v2f;
typedef __attribute__((ext_vector_type(8))) float v8f;

#define WAVES_PER_BLOCK 8
#define THREADS_PER_BLOCK (WAVES_PER_BLOCK * 32)

// ---------------------------------------------------------------------------
// Prep: pack each 3D point into (x, y, z, ||p||^2) so the hot loop does one
// aligned b128 load and zero arithmetic per point.
// ---------------------------------------------------------------------------
__global__ __launch_bounds__(256) void pack_points_kernel(
    const float* __restrict__ P,   // (count, 3)
    float4* __restrict__ out,      // (count)
    int count)
{
    int i = blockIdx.x * 256 + threadIdx.x;
    if (i < count) {
        const float* p = P + (size_t)i * 3;
        float x = p[0], y = p[1], z = p[2];
        out[i] = make_float4(x, y, z, x * x + y * y + z * z);
    }
}

// ---------------------------------------------------------------------------
// For each of the 16 "query" rows owned by a wave, compute
// min_j ||x_i - y_j||^2 over ALL target points with V_WMMA_F32_16X16X4_F32:
//   A = [-2x | 1] (16x4),  B = [y ; ||y||^2] (4x16),  C = 0
//   D = -2 x.y + ||y||^2 ;  full dist = D + ||x||^2 (added once per row at end,
//   legal because ||x||^2 is constant along the min axis).
// ---------------------------------------------------------------------------
__global__ __launch_bounds__(THREADS_PER_BLOCK) void chamfer_min_kernel(
    const float4* __restrict__ X4,    // (B, NX) packed query points (x,y,z,||x||^2)
    const float4* __restrict__ Y4,    // (B, NY) packed target points (y0,y1,y2,||y||^2)
    float* __restrict__ outMin,       // (B, NX) per-query min squared distance
    int NX, int NY)
{
    const int lane        = threadIdx.x & 31;
    const int waveInBlock = threadIdx.x >> 5;
    const int waveId      = blockIdx.x * WAVES_PER_BLOCK + waveInBlock;

    const int tilesPerBatch = NX >> 4;          // NX / 16
    const int b  = waveId / tilesPerBatch;
    const int n0 = (waveId % tilesPerBatch) << 4;

    const int half = lane >> 4;   // 0: lanes 0-15, 1: lanes 16-31
    const int l16  = lane & 15;

    const float4* Xb4 = X4 + (size_t)b * NX;
    const float4* Yb4 = Y4 + (size_t)b * NY;

    // ---- A-matrix fragment (16x4 f32, 2 VGPRs) ----
    // lanes 0-15 hold M=0..15 {VGPR0:K=0, VGPR1:K=1};
    // lanes 16-31 hold M=0..15 {VGPR0:K=2, VGPR1:K=3}.
    // A row m = [-2*x0, -2*x1, -2*x2, 1]
    const float4 xg = Xb4[n0 + l16];            // single aligned b128 load
    const float pn  = xg.w;                     // ||x||^2 for row l16

    v2f a;
    a.x = half ? (-2.0f * xg.z) : (-2.0f * xg.x);   // K=2 : K=0
    a.y = half ? 1.0f           : (-2.0f * xg.y);   // K=3 : K=1

    // C = 0: loop-invariant (inline-0 / hoisted zero regs)
    v8f czero;
#pragma unroll
    for (int i = 0; i < 8; ++i) czero[i] = 0.0f;

    v8f rowMin;
#pragma unroll
    for (int i = 0; i < 8; ++i) rowMin[i] = 3.0e38f;

    // ---- Sweep all target tiles ----
    const float4* yp = Yb4 + l16;               // this lane's column within tile
    for (int m0 = 0; m0 < NY; m0 += 16, yp += 16) {
        __builtin_prefetch(yp + 16, 0, 3);      // global_prefetch_b8, WGP scope

        const float4 g = *yp;                   // single aligned b128 load

        // B-matrix fragment (4x16 f32, 2 VGPRs):
        // lanes 0-15: {VGPR0:K=0, VGPR1:K=2}; lanes 16-31: {VGPR0:K=1, VGPR1:K=3};
        // column N = lane%16.  B col n = [y0, y1, y2, ||y||^2]^T
        v2f bb;
        bb.x = half ? g.y : g.x;   // K=1 : K=0
        bb.y = half ? g.w : g.z;   // K=3 : K=2

        // D = A*B + 0 = -2 x.y + ||y||^2 for the whole 16x16 tile
        v8f d = __builtin_amdgcn_wmma_f32_16x16x4_f32(
            /*neg_a=*/false, a, /*neg_b=*/false, bb,
            /*c_mod=*/(short)0, czero, /*reuse_a=*/false, /*reuse_b=*/false);

#pragma unroll
        for (int i = 0; i < 8; ++i) rowMin[i] = fminf(rowMin[i], d[i]);
    }

    // ---- Cross-lane min within each 16-lane half (row stays fixed per half) ----
#pragma unroll
    for (int off = 8; off >= 1; off >>= 1) {
#pragma unroll
        for (int i = 0; i < 8; ++i) {
            float o = __shfl_xor(rowMin[i], off, 32);
            rowMin[i] = fminf(rowMin[i], o);
        }
    }

    // Gather per-row ||x||^2 while ALL lanes are still active:
    // VGPR i of lanes 0-15 = row n0+i (pn held by lane i);
    // VGPR i of lanes 16-31 = row n0+8+i (pn held by lane 8+i).
    float pnRow[8];
#pragma unroll
    for (int i = 0; i < 8; ++i) pnRow[i] = __shfl(pn, half * 8 + i, 32);

    if (l16 == 0) {
        float* o = outMin + (size_t)b * NX + n0 + half * 8;
#pragma unroll
        for (int i = 0; i < 8; ++i)
            o[i] = fmaxf(rowMin[i] + pnRow[i], 0.0f);   // add ||x||^2, clamp
    }
}

// ---------------------------------------------------------------------------
// Single-block final reduction:
// loss = mean(minP) + mean(minG), means over (B*N) resp. (B*M)
// ---------------------------------------------------------------------------
__global__ __launch_bounds__(256) void chamfer_reduce_kernel(
    const float* __restrict__ minP, const float* __restrict__ minG,
    float* __restrict__ out, int cntP, int cntG)
{
    __shared__ float sm[256];
    const float invP = 1.0f / (float)cntP;
    const float invG = 1.0f / (float)cntG;
    float s = 0.0f;
    for (int i = threadIdx.x; i < cntP; i += 256) s += minP[i] * invP;
    for (int i = threadIdx.x; i < cntG; i += 256) s += minG[i] * invG;
    sm[threadIdx.x] = s;
    __syncthreads();
    for (int stride = 128; stride > 0; stride >>= 1) {
        if (threadIdx.x < stride) sm[threadIdx.x] += sm[threadIdx.x + stride];
        __syncthreads();
    }
    if (threadIdx.x == 0) out[0] = sm[0];
}

extern "C" void kernel_launch(void* const* d_in, const int* in_sizes, int n_in,
                              void* d_out, int out_size, void* d_ws, size_t ws_size,
                              hipStream_t stream) {
    const float* pred = (const float*)d_in[0];   // (B, N, 3) f32
    const float* gt   = (const float*)d_in[1];   // (B, M, 3) f32
    float* out = (float*)d_out;

    const int B = 8, D = 3;
    const int N = in_sizes[0] / (B * D);
    const int M = in_sizes[1] / (B * D);

    // Workspace layout (floats): minP[B*N] | minG[B*M] | packP[B*N*4] | packG[B*M*4]
    float*  minP  = (float*)d_ws;
    float*  minG  = minP + (size_t)B * N;
    float4* packP = (float4*)(minG + (size_t)B * M);   // 16B-aligned: (B*N+B*M)*4 bytes
    float4* packG = packP + (size_t)B * N;

    // Pack both point sets as (x, y, z, ||p||^2)
    {
        int cntP = B * N, cntG = B * M;
        pack_points_kernel<<<(cntP + 255) / 256, 256, 0, stream>>>(pred, packP, cntP);
        pack_points_kernel<<<(cntG + 255) / 256, 256, 0, stream>>>(gt,   packG, cntG);
    }

    // Pass 1: for each pred point, min squared distance to gt set
    {
        int waves  = (B * N) / 16;
        int blocks = waves / WAVES_PER_BLOCK;
        chamfer_min_kernel<<<blocks, THREADS_PER_BLOCK, 0, stream>>>(packP, packG, minP, N, M);
    }
    // Pass 2: roles swapped (distance tile formulation is symmetric)
    {
        int waves  = (B * M) / 16;
        int blocks = waves / WAVES_PER_BLOCK;
        chamfer_min_kernel<<<blocks, THREADS_PER_BLOCK, 0, stream>>>(packG, packP, minG, M, N);
    }
    // Final scalar
    chamfer_reduce_kernel<<<1, 256, 0, stream>>>(minP, minG, out, B * N, B * M);
}